// Signal_to_x_27848567947409
// MI455X (gfx1250) — compile-verified
//
#include <hip/hip_runtime.h>

// Pure bandwidth kernel: out = in, 8192*8192 fp32 (256 MiB each way).
// Roofline: 512 MiB / 23.3 TB/s ~= 23 us. Zero FLOPs -> no WMMA; the win is
// b128 non-temporal streaming (CDNA5 TH_NT cache policy, working set > 192MB L2).

typedef __attribute__((ext_vector_type(4))) float f32x4;

#define TPB 256            // 8 wave32s per block
#define VEC_PER_THREAD 4   // 4 x b128 = 64 B per thread

__global__ __launch_bounds__(TPB) void copy_stream_kernel(
    const f32x4* __restrict__ src, f32x4* __restrict__ dst, long long n4) {
  const long long blockBase = (long long)blockIdx.x * (TPB * VEC_PER_THREAD);
  const long long base = blockBase + threadIdx.x;

  if (blockBase + TPB * VEC_PER_THREAD <= n4) {
    // Fast path: whole block in range. Batch loads, then stores, so the
    // compiler keeps 4 global_load_b128 th:TH_LOAD_NT in flight per lane.
    f32x4 v0 = __builtin_nontemporal_load(&src[base + 0 * TPB]);
    f32x4 v1 = __builtin_nontemporal_load(&src[base + 1 * TPB]);
    f32x4 v2 = __builtin_nontemporal_load(&src[base + 2 * TPB]);
    f32x4 v3 = __builtin_nontemporal_load(&src[base + 3 * TPB]);
    __builtin_nontemporal_store(v0, &dst[base + 0 * TPB]);
    __builtin_nontemporal_store(v1, &dst[base + 1 * TPB]);
    __builtin_nontemporal_store(v2, &dst[base + 2 * TPB]);
    __builtin_nontemporal_store(v3, &dst[base + 3 * TPB]);
  } else {
    // Tail path (not taken for 8192x8192, kept for generality).
#pragma unroll
    for (int k = 0; k < VEC_PER_THREAD; ++k) {
      long long i = base + (long long)k * TPB;
      if (i < n4) {
        f32x4 v = __builtin_nontemporal_load(&src[i]);
        __builtin_nontemporal_store(v, &dst[i]);
      }
    }
  }
}

// Scalar cleanup for element counts not divisible by 4 (defensive; unused here).
__global__ __launch_bounds__(TPB) void copy_tail_kernel(
    const float* __restrict__ src, float* __restrict__ dst,
    long long start, long long n) {
  long long i = start + (long long)blockIdx.x * TPB + threadIdx.x;
  if (i < n) {
    dst[i] = __builtin_nontemporal_load(&src[i]);
  }
}

extern "C" void kernel_launch(void* const* d_in, const int* in_sizes, int n_in,
                              void* d_out, int out_size, void* d_ws, size_t ws_size,
                              hipStream_t stream) {
  (void)n_in; (void)d_ws; (void)ws_size; (void)out_size;
  const float* src = (const float*)d_in[0];
  float* dst = (float*)d_out;

  const long long n  = (long long)in_sizes[0];   // 67,108,864 floats
  const long long n4 = n >> 2;                   // 16,777,216 f32x4

  const long long vecsPerBlock = (long long)TPB * VEC_PER_THREAD;  // 1024
  const int blocks = (int)((n4 + vecsPerBlock - 1) / vecsPerBlock); // 16384

  if (blocks > 0) {
    copy_stream_kernel<<<blocks, TPB, 0, stream>>>(
        (const f32x4*)src, (f32x4*)dst, n4);
  }

  const long long rem = n - (n4 << 2);  // 0 for this problem
  if (rem > 0) {
    const int tblocks = (int)((rem + TPB - 1) / TPB);
    copy_tail_kernel<<<tblocks, TPB, 0, stream>>>(src, dst, n4 << 2, n);
  }
}